// KeypointNetwithIOLoss_8280696946693
// MI455X (gfx1250) — compile-verified
//
#include <hip/hip_runtime.h>
#include <hip/hip_bf16.h>
#include <math.h>

// ---------------------------------------------------------------------------
// KeypointNet IO loss on MI455X (gfx1250, wave32).
//   B=4 images, C=256 channels, H=48, W=64, N=H*W=3072 points.
//   Core: fused [N,C]x[C,N] similarity GEMM via V_WMMA_F32_16X16X4_F32 with
//   per-lane running masked argmax (replaces the reference's full argsort),
//   B-tiles staged block-cooperatively into LDS with async loads
//   (GLOBAL_LOAD_ASYNC_TO_LDS_B128, double-buffered, ASYNCcnt-synced).
// ---------------------------------------------------------------------------

typedef __attribute__((ext_vector_type(2))) float v2f;
typedef __attribute__((ext_vector_type(8))) float v8f;
typedef __attribute__((ext_vector_type(4))) int   v4i;

#if defined(__has_builtin)
# if __has_builtin(__builtin_amdgcn_global_load_async_to_lds_b128) && \
     __has_builtin(__builtin_amdgcn_s_wait_asynccnt)
#  define USE_ASYNC_LDS 1
# endif
#endif
#ifndef USE_ASYNC_LDS
# define USE_ASYNC_LDS 0
#endif

#if USE_ASYNC_LDS
// Builtin prototype (per hipcc diagnostic): (int4 AS1*, int4 AS3*, Ii, Ii)
typedef __attribute__((address_space(1))) v4i gbuf_t;   // global int4
typedef __attribute__((address_space(3))) v4i lbuf_t;   // LDS int4
#endif

#define Bimg 4
#define Cch  256
#define Hc   48
#define Wc   64
#define Npt  (Hc * Wc)      // 3072
#define NT   (Npt / 16)     // 192 16-wide tiles
#define MARGINv 0.2f
#define EPSv    1e-6f

__device__ __forceinline__ float waveReduceSum32(float v) {
#pragma unroll
  for (int off = 16; off; off >>= 1) v += __shfl_xor(v, off, 32);
  return v;
}

// ---------------------------------------------------------------------------
// Phase A: bilinear sample (align_corners=true, zero pad) + column L2 norm.
// One wave per point; lane handles channels c = lane + 32k (k=0..7).
// out layout: [B*N, C] row-major (1 KB rows -> GEMM friendly).
// ---------------------------------------------------------------------------
__global__ __launch_bounds__(256)
void sample_norm_kernel(const float* __restrict__ des,   // [B,C,H,W]
                        const float* __restrict__ pts,   // [B,H,W,2]
                        float* __restrict__ out) {       // [B*N, C]
  const int wave = threadIdx.x >> 5;
  const int lane = threadIdx.x & 31;
  const int g = blockIdx.x * 8 + wave;          // global point id, 0..B*N-1
  const int b = g / Npt;

  const float* im = des + (size_t)b * Cch * Npt;   // H*W == Npt
  const float px = pts[(size_t)g * 2 + 0];
  const float py = pts[(size_t)g * 2 + 1];
  const float x = (px + 1.0f) * 0.5f * (float)(Wc - 1);
  const float y = (py + 1.0f) * 0.5f * (float)(Hc - 1);
  const float x0f = floorf(x), y0f = floorf(y);
  const float wx = x - x0f, wy = y - y0f;
  const int x0 = (int)x0f, y0 = (int)y0f;
  const int x1 = x0 + 1, y1 = y0 + 1;
  const float vx0 = (x0 >= 0 && x0 < Wc) ? 1.f : 0.f;
  const float vx1 = (x1 >= 0 && x1 < Wc) ? 1.f : 0.f;
  const float vy0 = (y0 >= 0 && y0 < Hc) ? 1.f : 0.f;
  const float vy1 = (y1 >= 0 && y1 < Hc) ? 1.f : 0.f;
  const int x0c = min(max(x0, 0), Wc - 1), x1c = min(max(x1, 0), Wc - 1);
  const int y0c = min(max(y0, 0), Hc - 1), y1c = min(max(y1, 0), Hc - 1);
  const int o00 = y0c * Wc + x0c, o10 = y0c * Wc + x1c;
  const int o01 = y1c * Wc + x0c, o11 = y1c * Wc + x1c;
  const float w00 = (1.f - wx) * (1.f - wy) * vx0 * vy0;
  const float w10 = wx * (1.f - wy) * vx1 * vy0;
  const float w01 = (1.f - wx) * wy * vx0 * vy1;
  const float w11 = wx * wy * vx1 * vy1;

  float vals[8];
  float ss = 0.f;
#pragma unroll
  for (int k = 0; k < 8; ++k) {
    const float* ch = im + (size_t)(lane + 32 * k) * Npt;
    const float v = w00 * ch[o00] + w10 * ch[o10] + w01 * ch[o01] + w11 * ch[o11];
    vals[k] = v;
    ss += v * v;
  }
  ss = waveReduceSum32(ss);
  const float inv = 1.0f / sqrtf(ss);
  float* orow = out + (size_t)g * Cch;
#pragma unroll
  for (int k = 0; k < 8; ++k) orow[lane + 32 * k] = vals[k] * inv;
}

// ---------------------------------------------------------------------------
// Phase B: fused similarity GEMM (WMMA f32 16x16x4) + masked row argmax.
// One wave per 16-anchor tile; 8 waves/block share B-tiles via async LDS.
// A layout (16x4 f32): lanes 0-15 -> K {k,k+1}, lanes 16-31 -> K {k+2,k+3},
// row M = lane&15 (ISA 7.12.2). B mirrored (col N = lane&15).
// C/D: VGPR r holds rows r (lanes 0-15) and r+8 (lanes 16-31), col = lane&15.
// Argmax kept per-lane across all tiles; single cross-lane reduce at the end.
// ---------------------------------------------------------------------------
__global__ __launch_bounds__(256)
void sim_argmax_kernel(const float* __restrict__ ref,    // [B*N, C]
                       const float* __restrict__ tar,    // [B*N, C]
                       const float* __restrict__ tun,    // [B, 2, N]
                       const int* __restrict__ relax_p,
                       float* __restrict__ recall_cnt,   // [B]
                       int* __restrict__ neg_idx) {      // [B*N]
  const int wave = threadIdx.x >> 5;
  const int lane = threadIdx.x & 31;
  const int gw = blockIdx.x * 8 + wave;         // anchor-tile id over all images
  const int b = gw / NT;
  const int abase = (gw - b * NT) * 16;
  const int half = lane >> 4;
  const int l15 = lane & 15;
  const float relaxf = (float)relax_p[0];

  const float* R = ref + (size_t)b * Npt * Cch;
  const float* T = tar + (size_t)b * Npt * Cch;
  const float* tx = tun + (size_t)b * 2 * Npt;
  const float* ty = tx + Npt;

  const float* pA = R + (size_t)(abase + l15) * Cch + half * 2;
  const float axc = tx[abase + l15];            // this lane's anchor coords
  const float ayc = ty[abase + l15];

  // Hoisted per-row anchor coords: row (r + half*8) broadcast to this half.
  float axr[8], ayr[8];
#pragma unroll
  for (int r = 0; r < 8; ++r) {
    axr[r] = __shfl(axc, r + half * 8, 32);
    ayr[r] = __shfl(ayc, r + half * 8, 32);
  }

  float run_av[8]; int run_aj[8];               // per-lane best (sim, idx)
  float run_nv[8]; int run_nj[8];               // per-lane best valid negative
#pragma unroll
  for (int r = 0; r < 8; ++r) {
    run_av[r] = -1e30f; run_aj[r] = 0;
    run_nv[r] = -1e30f; run_nj[r] = 0;
  }

#if USE_ASYNC_LDS
  // Double-buffered block-cooperative B-tile staging: 2 x 16 rows x 256 f32.
  __shared__ float ldsT[2][16 * Cch];
  const int srow = threadIdx.x >> 4;            // 0..15: target row in tile
  const int scol = (threadIdx.x & 15) * 4;      // 16B chunk base (floats)
  auto issueTile = [&](int t, int buf) {
    const float* gbase = T + (size_t)(t * 16 + srow) * Cch + scol;
    float* lbase = &ldsT[buf][srow * Cch + scol];
#pragma unroll
    for (int p = 0; p < 4; ++p) {               // 4 x b128 per thread = 16KB/block
      __builtin_amdgcn_global_load_async_to_lds_b128(
          (gbuf_t*)(gbase + p * 64), (lbuf_t*)(lbase + p * 64), 0, 0);
    }
  };
  issueTile(0, 0);
#endif

  for (int t = 0; t < NT; ++t) {
    const int tbase = t * 16;
#if USE_ASYNC_LDS
    const int cur = t & 1;
    if (t + 1 < NT) {
      issueTile(t + 1, cur ^ 1);
      __builtin_amdgcn_s_wait_asynccnt(4);      // tile t's 4 ops done (in-order)
    } else {
      __builtin_amdgcn_s_wait_asynccnt(0);
    }
    __syncthreads();                            // all waves' slices of tile t in LDS
    const float* pB = &ldsT[cur][(size_t)l15 * Cch + half * 2];
#else
    const float* pB = T + (size_t)(tbase + l15) * Cch + half * 2;
#endif

    v8f acc0 = {};
    v8f acc1 = {};
#pragma unroll
    for (int k = 0; k < Cch; k += 8) {
      v2f a0 = *(const v2f*)(pA + k);
      v2f b0 = *(const v2f*)(pB + k);
      acc0 = __builtin_amdgcn_wmma_f32_16x16x4_f32(false, a0, false, b0,
                                                   (short)0, acc0, false, false);
      v2f a1 = *(const v2f*)(pA + k + 4);
      v2f b1 = *(const v2f*)(pB + k + 4);
      acc1 = __builtin_amdgcn_wmma_f32_16x16x4_f32(false, a1, false, b1,
                                                   (short)0, acc1, false, false);
    }
#if USE_ASYNC_LDS
    __syncthreads();                            // done reading buf before reuse
#endif

    const float jx = tx[tbase + l15];           // this lane's column coords
    const float jy = ty[tbase + l15];
    const int jcol = tbase + l15;

#pragma unroll
    for (int r = 0; r < 8; ++r) {
      const float v = acc0[r] + acc1[r];
      const bool correct = (fabsf(jx - axr[r]) <= relaxf) &&
                           (fabsf(jy - ayr[r]) <= relaxf);
      if (v > run_av[r]) { run_av[r] = v; run_aj[r] = jcol; }   // '>' keeps earliest
      const float nvv = correct ? -1e30f : v;
      if (nvv > run_nv[r]) { run_nv[r] = nvv; run_nj[r] = jcol; }
    }
  }

  // Single cross-lane argmax reduce per row (16-lane halves; xor<=8 stays in half).
#pragma unroll
  for (int r = 0; r < 8; ++r) {
    float bv = run_av[r]; int bj = run_aj[r];
    float nv = run_nv[r]; int nj = run_nj[r];
#pragma unroll
    for (int off = 8; off; off >>= 1) {
      const float ov = __shfl_xor(bv, off, 32);
      const int   oj = __shfl_xor(bj, off, 32);
      if (ov > bv || (ov == bv && oj < bj)) { bv = ov; bj = oj; }
      const float onv = __shfl_xor(nv, off, 32);
      const int   onj = __shfl_xor(nj, off, 32);
      if (onv > nv || (onv == nv && onj < nj)) { nv = onv; nj = onj; }
    }
    run_av[r] = bv; run_aj[r] = bj;
    run_nv[r] = nv; run_nj[r] = nj;
  }

  // Writers: lane 0 (rows 0..7) and lane 16 (rows 8..15).
  if (l15 == 0) {
    float cnt = 0.f;
#pragma unroll
    for (int r = 0; r < 8; ++r) {
      const int m = abase + r + half * 8;
      const int jb = run_aj[r];
      const float txm = tx[m], tym = ty[m];
      const float txj = tx[jb], tyj = ty[jb];
      cnt += (txj == txm && tyj == tym) ? 1.0f : 0.0f;  // exact-match recall
      neg_idx[b * Npt + m] = run_nj[r];
    }
    atomicAdd(&recall_cnt[b], cnt);
  }
}

// ---------------------------------------------------------------------------
// Phase C: triplet loss. One wave per anchor.
// ---------------------------------------------------------------------------
__global__ __launch_bounds__(256)
void triplet_loss_kernel(const float* __restrict__ ref,
                         const float* __restrict__ tar,
                         const int* __restrict__ neg_idx,
                         float* __restrict__ loss_sum) {
  const int wave = threadIdx.x >> 5;
  const int lane = threadIdx.x & 31;
  const int g = blockIdx.x * 8 + wave;        // anchor id over B*N
  const int b = g / Npt;
  const int j = neg_idx[g];
  const float* a = ref + (size_t)g * Cch;
  const float* p = tar + (size_t)g * Cch;
  const float* nrow = tar + ((size_t)b * Npt + j) * Cch;
  float sp = 0.f, sn = 0.f;
#pragma unroll
  for (int k = 0; k < 8; ++k) {
    const int c = lane + 32 * k;
    const float av = a[c];
    const float dp = av - p[c] + EPSv;    sp += dp * dp;
    const float dn = av - nrow[c] + EPSv; sn += dn * dn;
  }
  sp = waveReduceSum32(sp);
  sn = waveReduceSum32(sn);
  if (lane == 0) {
    const float l = sqrtf(sp) - sqrtf(sn) + MARGINv;
    atomicAdd(&loss_sum[b], fmaxf(l, 0.f));
  }
}

__global__ void init_kernel(float* loss_sum, float* recall_cnt) {
  if (threadIdx.x < Bimg) {
    loss_sum[threadIdx.x] = 0.f;
    recall_cnt[threadIdx.x] = 0.f;
  }
}

__global__ void final_kernel(const float* __restrict__ loss_sum,
                             const float* __restrict__ recall_cnt,
                             float* __restrict__ out) {
  if (threadIdx.x == 0) {
    float ls = 0.f, rc = 0.f;
    for (int b = 0; b < Bimg; ++b) { ls += loss_sum[b]; rc += recall_cnt[b]; }
    out[0] = ls / (float)(Bimg * Npt);
    out[1] = rc / (float)(Bimg * Npt);
  }
}

// ---------------------------------------------------------------------------
extern "C" void kernel_launch(void* const* d_in, const int* in_sizes, int n_in,
                              void* d_out, int out_size, void* d_ws, size_t ws_size,
                              hipStream_t stream) {
  const float* source_des    = (const float*)d_in[0];   // [B,C,H,W]
  const float* target_des    = (const float*)d_in[1];   // [B,C,H,W]
  const float* source_points = (const float*)d_in[2];   // [B,H,W,2]
  const float* tar_points    = (const float*)d_in[3];   // [B,H,W,2]
  const float* tar_points_un = (const float*)d_in[4];   // [B,2,H,W]
  const int*   relax_field   = (const int*)d_in[5];     // scalar
  (void)in_sizes; (void)n_in; (void)out_size; (void)ws_size;

  // Workspace layout (floats / ints, all 4-byte; base is 256B aligned).
  float* ref        = (float*)d_ws;                          // B*N*C
  float* tar        = ref + (size_t)Bimg * Npt * Cch;        // B*N*C
  float* loss_sum   = tar + (size_t)Bimg * Npt * Cch;        // B
  float* recall_cnt = loss_sum + Bimg;                       // B
  int*   neg        = (int*)(recall_cnt + Bimg);             // B*N

  init_kernel<<<1, 32, 0, stream>>>(loss_sum, recall_cnt);

  const int ptBlocks = (Bimg * Npt) / 8;                     // 1536
  sample_norm_kernel<<<ptBlocks, 256, 0, stream>>>(source_des, source_points, ref);
  sample_norm_kernel<<<ptBlocks, 256, 0, stream>>>(target_des, tar_points, tar);

  const int simBlocks = (Bimg * NT) / 8;                     // 96
  sim_argmax_kernel<<<simBlocks, 256, 0, stream>>>(ref, tar, tar_points_un,
                                                   relax_field, recall_cnt, neg);

  triplet_loss_kernel<<<ptBlocks, 256, 0, stream>>>(ref, tar, neg, loss_sum);

  final_kernel<<<1, 1, 0, stream>>>(loss_sum, recall_cnt, (float*)d_out);
}